// DAV_Block_Vectorized_76038101008621
// MI455X (gfx1250) — compile-verified
//
#include <hip/hip_runtime.h>
#include <math.h>

typedef __attribute__((ext_vector_type(2))) float v2f;
typedef __attribute__((ext_vector_type(4))) float v4f;
typedef __attribute__((ext_vector_type(8))) float v8f;

#define Bdim 2
#define Cdim 32
#define Hdim 96
#define Wdim 160
#define Ddim 32
#define NPIX (Bdim * Hdim * Wdim)          // 30720
#define CSTRIDE ((long)Hdim * Wdim * Ddim) // per-channel stride in warped_feat
#define QS_STRIDE 36                       // [w][c] LDS tile: 16B-aligned rows

// ---------------------------------------------------------------------------
// Pass 1: stream warped_feat (126 MB) exactly once; 2 FMAs per loaded float.
// Per pixel: s[d] = sum_c q[c]*k[c,d], e[d] = sum_c k[c,d]^2, a = sum_c q[c]^2.
// Block = 256 threads = 8 waves, 32 consecutive-w pixels of one (b,h) row.
// Lane = (pixel-in-wave, d-quad): b128 k loads, b128 q loads from LDS.
// ---------------------------------------------------------------------------
__global__ __launch_bounds__(256) void dav_pass1(
    const float* __restrict__ q, const float* __restrict__ wf,
    float* __restrict__ S, float* __restrict__ E, float* __restrict__ Aq) {
  __shared__ float qs[32 * QS_STRIDE]; // qs[w][c], transposed for b128 reads

  const int t   = threadIdx.x;
  const int blk = blockIdx.x; // (b*96 + h)*5 + wq
  const int wq  = blk % 5;
  const int h   = (blk / 5) % Hdim;
  const int b   = blk / (5 * Hdim);
  const int w0  = wq * 32;

  // Stage q tile transposed: global coalesced along w, LDS write [w][c].
  {
    const int wl = t & 31;
    const int c0 = t >> 5;
#pragma unroll
    for (int r = 0; r < 4; ++r) {
      const int c = c0 + 8 * r;
      qs[wl * QS_STRIDE + c] = q[((b * Cdim + c) * Hdim + h) * Wdim + w0 + wl];
    }
  }
  __syncthreads();

  // a[p] = sum_c q^2 : 32 lanes, one pixel each.
  if (t < 32) {
    float acc = 0.f;
#pragma unroll
    for (int c = 0; c < Cdim; ++c) {
      const float v = qs[t * QS_STRIDE + c];
      acc += v * v;
    }
    Aq[(b * Hdim + h) * Wdim + w0 + t] = acc;
  }

  const int wave = t >> 5;
  const int lane = t & 31;
  const int pxl  = wave * 4 + (lane >> 3); // 0..31 within the 32-pixel tile
  const int d4   = lane & 7;               // d-quad
  const int w    = w0 + pxl;

  const float* kp =
      wf + ((long)(b * Cdim) * (Hdim * Wdim) + (long)h * Wdim + w) * Ddim +
      d4 * 4;

  v4f s4 = {0.f, 0.f, 0.f, 0.f};
  v4f e4 = {0.f, 0.f, 0.f, 0.f};
#pragma unroll
  for (int c4 = 0; c4 < 8; ++c4) {
    const v4f qv4 = *(const v4f*)&qs[pxl * QS_STRIDE + c4 * 4]; // ds_load_b128
#pragma unroll
    for (int u = 0; u < 4; ++u) {
      const v4f kv = *(const v4f*)kp; // global_load_b128, fully coalesced
      kp += CSTRIDE;
      const float qv = qv4[u];
      const v4f qvv = {qv, qv, qv, qv};
      s4 += qvv * kv;
      e4 += kv * kv;
    }
  }

  const long pix = (long)(b * Hdim + h) * Wdim + w;
  *(v4f*)(S + pix * Ddim + d4 * 4) = s4;
  *(v4f*)(E + pix * Ddim + d4 * 4) = e4;
}

// ---------------------------------------------------------------------------
// Pass 2: 3x3 zero-padded box sum + normalization on L2-resident data.
// Horizontal 3-tap filter = banded matmul via V_WMMA_F32_16X16X4_F32:
//   Out[d, w0+n] = sum_u V[d, w0-1+u] * T[u, n],  T[u,n] = 1{|u-1-n|<=1}
// Column validity is folded into T (B operand) so A loads are unconditional
// from clamped addresses; row validity = clamped-offset loads * {0,1} weight.
// One wave per (b, h, d-half, w-tile); fully branch-free, EXEC always full.
// A 16x4 f32 layout: [m,k] -> lane = (m&15)+16*(k>>1), vgpr = k&1.
// B 4x16 mirrors it: [k,n] -> lane = (n&15)+16*(k>>1), vgpr = k&1.
// ---------------------------------------------------------------------------
__global__ __launch_bounds__(128) void dav_pass2(
    const float* __restrict__ S, const float* __restrict__ E,
    const float* __restrict__ Aq, float* __restrict__ out) {
  const int wid  = blockIdx.x * 4 + (threadIdx.x >> 5);
  const int lane = threadIdx.x & 31;

  const int n     = wid % 10;       // w-tile
  const int dh    = (wid / 10) & 1; // d-half
  const int h     = (wid / 20) % Hdim;
  const int b     = wid / (20 * Hdim);
  const int w0    = n * 16;
  const int dbase = dh * 16;

  const int nn   = lane & 15; // = m for A rows, = n for B columns
  const int half = lane >> 4;
  const int col  = dbase + nn;

  // Vertical 3-sum: clamped offsets + 0/1 weights (no branches).
  const int  rowStride = Wdim * Ddim;
  const float gU = (h > 0) ? 1.f : 0.f;
  const float gD = (h < Hdim - 1) ? 1.f : 0.f;
  const int  oU = (h > 0) ? -rowStride : 0;
  const int  oD = (h < Hdim - 1) ? rowStride : 0;

  const long rowbase = (long)(b * Hdim + h) * Wdim * Ddim + col;
  const float* Srow = S + rowbase;
  const float* Erow = E + rowbase;

  v8f accS = {0.f, 0.f, 0.f, 0.f, 0.f, 0.f, 0.f, 0.f};
  v8f accE = {0.f, 0.f, 0.f, 0.f, 0.f, 0.f, 0.f, 0.f};

#pragma unroll
  for (int j = 0; j < 5; ++j) {
    v2f aS, aE, bb;
#pragma unroll
    for (int v = 0; v < 2; ++v) {
      const int k   = v + 2 * half;       // K index held by this lane/vgpr
      const int wp  = w0 - 1 + 4 * j + k; // source column w'
      const int wpc = wp < 0 ? 0 : (wp > Wdim - 1 ? Wdim - 1 : wp);
      const long idx = (long)wpc * Ddim;

      const float s1 = Srow[idx];
      const float s0 = Srow[idx + oU];
      const float s2 = Srow[idx + oD];
      const float e1 = Erow[idx];
      const float e0 = Erow[idx + oU];
      const float e2 = Erow[idx + oD];
      aS[v] = fmaf(gU, s0, fmaf(gD, s2, s1));
      aE[v] = fmaf(gU, e0, fmaf(gD, e2, e1));

      // Band entry AND column validity both live in B.
      const int diff = 4 * j + k - 1 - nn;
      const bool inband = (unsigned)(diff + 1) <= 2u;
      const bool wok    = (unsigned)wp < (unsigned)Wdim;
      bb[v] = (inband && wok) ? 1.f : 0.f;
    }
    // D = A x B + C  (fp32, exact accumulate). EXEC is all-ones here.
    accS = __builtin_amdgcn_wmma_f32_16x16x4_f32(false, aS, false, bb,
                                                 (short)0, accS, false, false);
    accE = __builtin_amdgcn_wmma_f32_16x16x4_f32(false, aE, false, bb,
                                                 (short)0, accE, false, false);
  }

  // nq for this lane's output column: 3x3 box sum of a (clamp + weight).
  const int wcol = w0 + nn;
  const float* Arow = Aq + (b * Hdim + h) * Wdim;
  float bsa = 0.f;
#pragma unroll
  for (int dy = -1; dy <= 1; ++dy) {
    const int hh = h + dy;
    const bool hok = (unsigned)hh < (unsigned)Hdim;
    const int ho = hok ? dy * Wdim : 0;
    const float gh = hok ? 1.f : 0.f;
#pragma unroll
    for (int dx = -1; dx <= 1; ++dx) {
      const int ww  = wcol + dx;
      const bool wok = (unsigned)ww < (unsigned)Wdim;
      const int wwc = ww < 0 ? 0 : (ww > Wdim - 1 ? Wdim - 1 : ww);
      const float g = wok ? gh : 0.f;
      bsa = fmaf(g, Arow[ho + wwc], bsa);
    }
  }
  const float nq = fmaxf(sqrtf(bsa), 1e-12f);

  // C/D layout: VGPR r holds M = r (lanes 0-15) / r+8 (lanes 16-31), N = nn.
#pragma unroll
  for (int r = 0; r < 8; ++r) {
    const int d     = dbase + r + 8 * half;
    const float nk  = fmaxf(sqrtf(accE[r]), 1e-12f);
    const float sim = accS[r] / (nq * nk);
    out[((long)(b * Ddim + d) * Hdim + h) * Wdim + wcol] = sim;
  }
}

extern "C" void kernel_launch(void* const* d_in, const int* in_sizes, int n_in,
                              void* d_out, int out_size, void* d_ws,
                              size_t ws_size, hipStream_t stream) {
  const float* q  = (const float*)d_in[0];
  const float* wf = (const float*)d_in[1];
  float* out = (float*)d_out;

  // Workspace layout: S (3.93 MB) | E (3.93 MB) | Aq (0.12 MB)  ~= 8 MB.
  float* S  = (float*)d_ws;
  float* E  = S + (size_t)NPIX * Ddim;
  float* Aq = E + (size_t)NPIX * Ddim;

  dav_pass1<<<Bdim * Hdim * 5, 256, 0, stream>>>(q, wf, S, E, Aq);
  dav_pass2<<<(Bdim * Hdim * 2 * 10) / 4, 128, 0, stream>>>(S, E, Aq, out);
}